// MultiHeadedAttention_43258910605364
// MI455X (gfx1250) — compile-verified
//
#include <hip/hip_runtime.h>

// Problem constants (from reference)
#define A_  16
#define B_  16
#define S_  128
#define D_  512
#define H_  8
#define DK  64
#define L_  5
#define N_  256          // A_*B_
#define TQ  132          // S_ + L_ - 1

typedef float v2f __attribute__((ext_vector_type(2)));
typedef float v8f __attribute__((ext_vector_type(8)));

__device__ __forceinline__ v8f wmma_f32(v2f a, v2f b, v8f c) {
  // D = A(16x4 f32) x B(4x16 f32) + C(16x16 f32)
  return __builtin_amdgcn_wmma_f32_16x16x4_f32(false, a, false, b, (short)0, c,
                                               false, false);
}

// ---------------------------------------------------------------------------
// Batched projection GEMM:  Y[n, r, o] = sum_d Xrow(n,r)[d] * W[o, d] + bias[o]
// Xrow(n,r) = (r < padRows) ? pad[n, r, :] : x[n, r - padRows, :]
// One wave computes a 16x64 strip (4 accumulators share one A fragment).
// Software-pipelined: next K-step's 5 fragments are loaded before the current
// step's 4 independent v_wmma ops, so loads overlap matrix math.
// ---------------------------------------------------------------------------
__global__ __launch_bounds__(256)
void wmma_proj_kernel(const float* __restrict__ x,
                      const float* __restrict__ pad,
                      const float* __restrict__ W,
                      const float* __restrict__ bias,
                      float* __restrict__ y,
                      int rows, int padRows)
{
  const int n    = blockIdx.y;
  const int wave = threadIdx.x >> 5;
  const int lane = threadIdx.x & 31;
  const int NG   = 8;                        // 32 ntiles / 4 per wave
  const int unit = blockIdx.x * 8 + wave;    // uniform per wave; grid.x==tilesM
  const int mt   = unit / NG;
  const int ng   = unit - mt * NG;           // handles ntiles ng*4 .. ng*4+3

  const int halfK = lane >> 4;               // 0/1 -> K pair {0,1} / {2,3}
  const int l15   = lane & 15;
  const int ko    = 2 * halfK;

  const int row  = mt * 16 + l15;
  const int rowc = row < rows ? row : rows - 1;      // clamp: safe address
  const int xrows = rows - padRows;
  const float* arow = (rowc < padRows)
       ? pad + ((size_t)n * padRows + rowc) * D_
       : x   + ((size_t)n * xrows + (rowc - padRows)) * D_;

  const int o0 = ng * 64 + l15;
  const float* bcol0 = W + (size_t)(o0)      * D_;
  const float* bcol1 = W + (size_t)(o0 + 16) * D_;
  const float* bcol2 = W + (size_t)(o0 + 32) * D_;
  const float* bcol3 = W + (size_t)(o0 + 48) * D_;

  __builtin_prefetch(arow, 0, 3);
  __builtin_prefetch(bcol0, 0, 3);

  v8f acc0, acc1, acc2, acc3;
  {
    const float b0 = bias[o0], b1 = bias[o0 + 16];
    const float b2 = bias[o0 + 32], b3 = bias[o0 + 48];
#pragma unroll
    for (int i = 0; i < 8; ++i) { acc0[i] = b0; acc1[i] = b1; acc2[i] = b2; acc3[i] = b3; }
  }

  // Software pipeline: prologue loads, then loop {prefetch next, 4x wmma}.
  v2f a_c  = *(const v2f*)(arow  + ko);
  v2f b0_c = *(const v2f*)(bcol0 + ko);
  v2f b1_c = *(const v2f*)(bcol1 + ko);
  v2f b2_c = *(const v2f*)(bcol2 + ko);
  v2f b3_c = *(const v2f*)(bcol3 + ko);
#pragma unroll 4
  for (int k = 0; k < D_; k += 4) {
    const int kn = ((k + 4) & (D_ - 1)) + ko;   // wrapped prefetch (last iter dummy)
    const v2f a_n  = *(const v2f*)(arow  + kn);
    const v2f b0_n = *(const v2f*)(bcol0 + kn);
    const v2f b1_n = *(const v2f*)(bcol1 + kn);
    const v2f b2_n = *(const v2f*)(bcol2 + kn);
    const v2f b3_n = *(const v2f*)(bcol3 + kn);
    acc0 = wmma_f32(a_c, b0_c, acc0);
    acc1 = wmma_f32(a_c, b1_c, acc1);
    acc2 = wmma_f32(a_c, b2_c, acc2);
    acc3 = wmma_f32(a_c, b3_c, acc3);
    a_c = a_n; b0_c = b0_n; b1_c = b1_n; b2_c = b2_n; b3_c = b3_n;
  }

  float* yb = y + (size_t)n * rows * D_;
#pragma unroll
  for (int r = 0; r < 8; ++r) {
    const int m = mt * 16 + r + 8 * halfK;   // C layout: VGPR r -> M=r / r+8
    if (m < rows) {
      float* yr = yb + (size_t)m * D_;
      yr[o0]      = acc0[r];
      yr[o0 + 16] = acc1[r];
      yr[o0 + 32] = acc2[r];
      yr[o0 + 48] = acc3[r];
    }
  }
}

// ---------------------------------------------------------------------------
// Sliding-window softmax weights: w[n, s, l] = softmax_l(center . win_l / sqrt(D))
// center = y[n, s+L-1, :], win_l = y[n, s+l, :].  One wave per (n, s).
// ---------------------------------------------------------------------------
__global__ __launch_bounds__(256)
void win_weights_kernel(const float* __restrict__ y, float* __restrict__ w)
{
  const int n    = blockIdx.x;
  const int wave = threadIdx.x >> 5;
  const int lane = threadIdx.x & 31;
  const float invSqrtD = 0.044194173824159216f;   // 1/sqrt(512)
  const float* yb = y + (size_t)n * TQ * D_;

  for (int s = wave; s < S_; s += 8) {
    const float* c = yb + (size_t)(s + L_ - 1) * D_ + lane * 16;
    float dots[L_];
#pragma unroll
    for (int l = 0; l < L_; ++l) {
      const float* wn = yb + (size_t)(s + l) * D_ + lane * 16;
      float d = 0.f;
#pragma unroll
      for (int j = 0; j < 16; ++j) d += c[j] * wn[j];
      dots[l] = d;
    }
#pragma unroll
    for (int off = 16; off >= 1; off >>= 1) {
#pragma unroll
      for (int l = 0; l < L_; ++l) dots[l] += __shfl_xor(dots[l], off, 32);
    }
    float sc[L_];
#pragma unroll
    for (int l = 0; l < L_; ++l) sc[l] = dots[l] * invSqrtD;
    float mx = sc[0];
#pragma unroll
    for (int l = 1; l < L_; ++l) mx = fmaxf(mx, sc[l]);
    float sum = 0.f;
#pragma unroll
    for (int l = 0; l < L_; ++l) { sc[l] = __expf(sc[l] - mx); sum += sc[l]; }
    const float inv = 1.f / sum;
    if (lane == 0) {
      float* wo = w + ((size_t)n * S_ + s) * L_;
#pragma unroll
      for (int l = 0; l < L_; ++l) wo[l] = sc[l] * inv;
    }
  }
}

// ---------------------------------------------------------------------------
// Scrambled window combine (replicates the reference's reshape exactly):
//   out[n,s',d] = sum_{j=0..4} w[n, f/5, f%5] * y[n, f/5 + f%5, d],  f = s'+128j
// ---------------------------------------------------------------------------
__global__ __launch_bounds__(128)
void win_combine_kernel(const float* __restrict__ y, const float* __restrict__ w,
                        float* __restrict__ out)
{
  const int n  = blockIdx.y;
  const int sp = blockIdx.x;
  const int d  = threadIdx.x * 4;
  const float* yb = y + (size_t)n * TQ * D_;
  const float* wb = w + (size_t)n * S_ * L_;
  float4 acc = make_float4(0.f, 0.f, 0.f, 0.f);
#pragma unroll
  for (int j = 0; j < L_; ++j) {
    const int f  = sp + S_ * j;
    const int sj = f / L_;
    const int lj = f - sj * L_;
    const float  sc = wb[sj * L_ + lj];
    const float4 v  = *(const float4*)(yb + (size_t)(sj + lj) * D_ + d);
    acc.x += sc * v.x; acc.y += sc * v.y; acc.z += sc * v.z; acc.w += sc * v.w;
  }
  *(float4*)(out + ((size_t)n * S_ + sp) * D_ + d) = acc;
}

// ---------------------------------------------------------------------------
// Attention over S: one block per (n, head).
//  - K and V for the head are staged once into LDS (V transposed so the P.V
//    B-fragment becomes one aligned ds_load_b64).
//  - Scores staged in LDS in two 64-row chunks, row softmax, then P.V with the
//    LDS A-fragment feeding 2 accumulators per K-step.
// LDS: P 33.8KB + KS 34.8KB + VT 33.3KB = ~102KB (WGP budget 320KB).
// ---------------------------------------------------------------------------
__global__ __launch_bounds__(256)
void attn_s_kernel(const float* __restrict__ q, const float* __restrict__ k,
                   const float* __restrict__ v, float* __restrict__ o)
{
  __shared__ float P[64 * 132];      // score chunk, stride 132
  __shared__ float KS[128 * 68];     // K[t][d], stride 68
  __shared__ float VT[64 * 130];     // V^T[d][t], stride 130
  const int STR  = 132;
  const int n    = blockIdx.x >> 3;
  const int h    = blockIdx.x & 7;
  const int wave = threadIdx.x >> 5;
  const int lane = threadIdx.x & 31;
  const int halfK = lane >> 4;
  const int l15   = lane & 15;
  const int ko    = 2 * halfK;
  const int hd    = h * DK;

  const float* qb = q + (size_t)n * S_ * D_ + hd;
  const float* kb = k + (size_t)n * S_ * D_ + hd;
  const float* vb = v + (size_t)n * S_ * D_ + hd;
  float*       ob = o + (size_t)n * S_ * D_ + hd;

  // Stage K (row-major) and V (transposed) into LDS cooperatively.
  for (int i = threadIdx.x; i < S_ * 16; i += 256) {
    const int t  = i >> 4;
    const int c4 = (i & 15) * 4;
    const float4 kv = *(const float4*)(kb + (size_t)t * D_ + c4);
    *(float4*)&KS[t * 68 + c4] = kv;
    const float4 vv = *(const float4*)(vb + (size_t)t * D_ + c4);
    VT[(c4 + 0) * 130 + t] = vv.x;
    VT[(c4 + 1) * 130 + t] = vv.y;
    VT[(c4 + 2) * 130 + t] = vv.z;
    VT[(c4 + 3) * 130 + t] = vv.w;
  }
  __syncthreads();

  for (int chunk = 0; chunk < 2; ++chunk) {
    const int mbase = chunk * 64;
    // Phase 1: S = Q.K^T / 8.  Wave -> (mtile = wave>>1, nhalf = wave&1):
    // one pipelined global A fragment feeds 4 accumulators (B from LDS).
    {
      const int lmt = wave >> 1;
      const int nh  = wave & 1;
      const float* arow = qb + (size_t)(mbase + lmt * 16 + l15) * D_;
      const float* k0 = &KS[(nh * 64 +  0 + l15) * 68 + ko];
      const float* k1 = &KS[(nh * 64 + 16 + l15) * 68 + ko];
      const float* k2 = &KS[(nh * 64 + 32 + l15) * 68 + ko];
      const float* k3 = &KS[(nh * 64 + 48 + l15) * 68 + ko];
      v8f a0 = {}, a1 = {}, a2 = {}, a3 = {};
      v2f a_c = *(const v2f*)(arow + ko);
#pragma unroll 4
      for (int kk = 0; kk < DK; kk += 4) {
        const int kn = ((kk + 4) & (DK - 1)) + ko;
        const v2f a_n = *(const v2f*)(arow + kn);
        a0 = wmma_f32(a_c, *(const v2f*)(k0 + kk), a0);
        a1 = wmma_f32(a_c, *(const v2f*)(k1 + kk), a1);
        a2 = wmma_f32(a_c, *(const v2f*)(k2 + kk), a2);
        a3 = wmma_f32(a_c, *(const v2f*)(k3 + kk), a3);
        a_c = a_n;
      }
#pragma unroll
      for (int r = 0; r < 8; ++r) {
        const int lm = lmt * 16 + r + 8 * halfK;
        float* pr = &P[lm * STR + nh * 64 + l15];
        pr[0]  = a0[r] * 0.125f;
        pr[16] = a1[r] * 0.125f;
        pr[32] = a2[r] * 0.125f;
        pr[48] = a3[r] * 0.125f;
      }
    }
    __syncthreads();
    // Phase 2: row softmax (one thread per row)
    if (threadIdx.x < 64) {
      float* rowp = &P[threadIdx.x * STR];
      float mx = -3.0e38f;
      for (int j = 0; j < S_; ++j) mx = fmaxf(mx, rowp[j]);
      float sum = 0.f;
      for (int j = 0; j < S_; ++j) { const float e = __expf(rowp[j] - mx); rowp[j] = e; sum += e; }
      const float inv = 1.f / sum;
      for (int j = 0; j < S_; ++j) rowp[j] *= inv;
    }
    __syncthreads();
    // Phase 3: O = P.V, all fragments from LDS.
    {
      const int lmt = wave >> 1;
      const int np  = wave & 1;
      const int oc0 = np * 32 + l15;
      const float* prow = &P[(lmt * 16 + l15) * STR + ko];
      const float* vt0  = &VT[(size_t)oc0 * 130 + ko];
      const float* vt1  = vt0 + 16 * 130;
      v8f c0 = {}, c1 = {};
#pragma unroll 8
      for (int kk = 0; kk < S_; kk += 4) {
        const v2f a = *(const v2f*)(prow + kk);
        c0 = wmma_f32(a, *(const v2f*)(vt0 + kk), c0);
        c1 = wmma_f32(a, *(const v2f*)(vt1 + kk), c1);
      }
#pragma unroll
      for (int r = 0; r < 8; ++r) {
        const int m = mbase + lmt * 16 + r + 8 * halfK;
        ob[(size_t)m * D_ + oc0]      = c0[r];
        ob[(size_t)m * D_ + oc0 + 16] = c1[r];
      }
    }
    __syncthreads();
  }
}

// ---------------------------------------------------------------------------
// Attention over A: one block per n2 = s*B + b; one wave per head.
// 16x16 scores kept in accumulators, softmax via shfl_xor within 16-lane
// halves (matches C layout), P re-laid-out through a small LDS tile; the P
// fragment then feeds 4 accumulators per K-step for O = P.V.
// ---------------------------------------------------------------------------
__global__ __launch_bounds__(256)
void attn_a_kernel(const float* __restrict__ x1, float* __restrict__ x2)
{
  __shared__ float P[H_][16][18];
  const int n2   = blockIdx.x;
  const int s    = n2 >> 4;        // n2 / B_
  const int bb   = n2 & 15;        // n2 % B_
  const int h    = threadIdx.x >> 5;
  const int lane = threadIdx.x & 31;
  const int halfK = lane >> 4;
  const int l15   = lane & 15;
  const int ko    = 2 * halfK;
  const size_t tokStride = (size_t)B_ * S_ * D_;   // stride between a tokens
  const float* base = x1 + ((size_t)bb * S_ + s) * D_ + h * DK;

  // Scores: 16x16 = X.X^T over K=64 (Q==K, so A and B fragments coincide)
  v8f acc = {};
  {
    const float* arow = base + (size_t)l15 * tokStride;
#pragma unroll
    for (int kk = 0; kk < DK; kk += 4) {
      const v2f a = *(const v2f*)(arow + kk + ko);
      acc = wmma_f32(a, a, acc);
    }
  }
  // Softmax per row (row M lives entirely in one 16-lane half)
  float pr[8];
#pragma unroll
  for (int r = 0; r < 8; ++r) pr[r] = acc[r] * 0.125f;
#pragma unroll
  for (int r = 0; r < 8; ++r) {
    float mx = pr[r];
#pragma unroll
    for (int off = 8; off >= 1; off >>= 1) mx = fmaxf(mx, __shfl_xor(mx, off, 32));
    const float e = __expf(pr[r] - mx);
    float sum = e;
#pragma unroll
    for (int off = 8; off >= 1; off >>= 1) sum += __shfl_xor(sum, off, 32);
    pr[r] = e / sum;
  }
#pragma unroll
  for (int r = 0; r < 8; ++r) P[h][r + 8 * halfK][l15] = pr[r];
  __syncthreads();

  // O = P.V : M=16, N=64, K=16; one LDS A fragment feeds 4 accumulators.
  v8f o0 = {}, o1 = {}, o2 = {}, o3 = {};
#pragma unroll
  for (int kk = 0; kk < 16; kk += 4) {
    const int kt = kk + ko;
    const v2f a = *(const v2f*)(&P[h][l15][kt]);
    const float* r0 = base + (size_t)kt * tokStride;
    const float* r1 = base + (size_t)(kt + 1) * tokStride;
    v2f b;
    b.x = r0[l15];      b.y = r1[l15];      o0 = wmma_f32(a, b, o0);
    b.x = r0[16 + l15]; b.y = r1[16 + l15]; o1 = wmma_f32(a, b, o1);
    b.x = r0[32 + l15]; b.y = r1[32 + l15]; o2 = wmma_f32(a, b, o2);
    b.x = r0[48 + l15]; b.y = r1[48 + l15]; o3 = wmma_f32(a, b, o3);
  }
#pragma unroll
  for (int r = 0; r < 8; ++r) {
    const int arow = r + 8 * halfK;          // query token index a
    float* xr = x2 + ((size_t)arow * B_ + bb) * S_ * D_ + (size_t)s * D_ + h * DK + l15;
    xr[0]  = o0[r];
    xr[16] = o1[r];
    xr[32] = o2[r];
    xr[48] = o3[r];
  }
}

// ---------------------------------------------------------------------------
extern "C" void kernel_launch(void* const* d_in, const int* in_sizes, int n_in,
                              void* d_out, int out_size, void* d_ws, size_t ws_size,
                              hipStream_t stream)
{
  (void)in_sizes; (void)n_in; (void)out_size; (void)ws_size;
  const float* query = (const float*)d_in[0];
  const float* key   = (const float*)d_in[1];
  const float* value = (const float*)d_in[2];
  /* d_in[3] = mask (all ones, unused by reference) */
  const float* padq  = (const float*)d_in[4];
  const float* padk  = (const float*)d_in[5];
  const float* Wq    = (const float*)d_in[6];
  const float* bq    = (const float*)d_in[7];
  const float* Wk    = (const float*)d_in[8];
  const float* bk    = (const float*)d_in[9];
  const float* Wv    = (const float*)d_in[10];
  const float* bv    = (const float*)d_in[11];
  const float* Wo    = (const float*)d_in[12];
  const float* bo    = (const float*)d_in[13];
  float* out = (float*)d_out;

  float* ws = (float*)d_ws;
  size_t off = 0;
  float* YQ  = ws + off; off += (size_t)N_ * TQ * D_;   // 17.3M f32
  float* YK  = ws + off; off += (size_t)N_ * TQ * D_;
  float* QP  = ws + off; off += (size_t)N_ * S_ * D_;
  float* KP  = ws + off; off += (size_t)N_ * S_ * D_;
  float* VP  = ws + off; off += (size_t)N_ * S_ * D_;
  float* WQW = ws + off; off += (size_t)N_ * S_ * L_;
  float* WKW = ws + off; off += (size_t)N_ * S_ * L_;
  float* X1  = YQ;   // YQ dead after win_combine(Q)
  float* X2  = YK;   // YK dead after win_combine(K)

  const dim3 blk256(256), blk128(128);

  // 1) Projections. Q/K: 132 rows (pad||x): 9 mtiles x 8 wave-units per block.
  wmma_proj_kernel<<<dim3(9, N_), blk256, 0, stream>>>(query, padq, Wq, bq, YQ, TQ, L_ - 1);
  wmma_proj_kernel<<<dim3(9, N_), blk256, 0, stream>>>(key,   padk, Wk, bk, YK, TQ, L_ - 1);
  wmma_proj_kernel<<<dim3(8, N_), blk256, 0, stream>>>(value, nullptr, Wv, bv, VP, S_, 0);

  // 2) Local sliding-window reweighting for q and k.
  win_weights_kernel<<<dim3(N_), blk256, 0, stream>>>(YQ, WQW);
  win_weights_kernel<<<dim3(N_), blk256, 0, stream>>>(YK, WKW);
  win_combine_kernel<<<dim3(S_, N_), blk128, 0, stream>>>(YQ, WQW, QP);
  win_combine_kernel<<<dim3(S_, N_), blk128, 0, stream>>>(YK, WKW, KP);

  // 3) Attention over S (one block per (n, head)).
  attn_s_kernel<<<dim3(N_ * H_), blk256, 0, stream>>>(QP, KP, VP, X1);

  // 4) Attention over A (one block per s*B+b, one wave per head).
  attn_a_kernel<<<dim3(S_ * B_), blk256, 0, stream>>>(X1, X2);

  // 5) Output projection into d_out ([A,B,S,D] flat == [n,s,o]).
  wmma_proj_kernel<<<dim3(8, N_), blk256, 0, stream>>>(X2, nullptr, Wo, bo, out, S_, 0);
}